// MultiHeadedAttentionCat_36352603194094
// MI455X (gfx1250) — compile-verified
//
#include <hip/hip_runtime.h>
#include <math.h>

// ---------------------------------------------------------------------------
// MultiHeadedAttentionCat for MI455X (gfx1250), f32 end-to-end via
// V_WMMA_F32_16X16X4_F32 + async LDS staging (ASYNCcnt) for K/V tiles.
// B=1, N=4096, D_MODEL=256, H=4, DIM=64.
// ---------------------------------------------------------------------------

typedef float v2f __attribute__((ext_vector_type(2)));
typedef float v8f __attribute__((ext_vector_type(8)));

#define N_TOK 4096
#define NH    4
#define DHEAD 64
#define VDIM  80      // 67 value dims (64 + 3 coords) padded to 5*16
#define DM    256
#define DOUT  284     // d_model + 28
#define KB    32      // keys staged per block iteration

__device__ __forceinline__ v8f wmma4(v2f a, v2f b, v8f c) {
  // D = A(16x4 f32) * B(4x16 f32) + C(16x16 f32)
  return __builtin_amdgcn_wmma_f32_16x16x4_f32(false, a, false, b, (short)0, c,
                                               false, false);
}

// Async DMA: 16 bytes global -> LDS per lane, tracked by ASYNCcnt.
__device__ __forceinline__ void async_ld_b128(unsigned lds_byte_off,
                                              const void* gaddr) {
  asm volatile("global_load_async_to_lds_b128 %0, %1, off"
               :: "v"(lds_byte_off), "v"(gaddr)
               : "memory");
}
__device__ __forceinline__ void wait_async0() {
  asm volatile("s_wait_asynccnt 0x0" ::: "memory");
}
// Low 32 bits of a generic shared-aperture address are the LDS byte offset.
__device__ __forceinline__ unsigned lds_off(const void* p) {
  return (unsigned)(size_t)p;
}

// ---------------------------------------------------------------------------
// Kernel 1: projections  P = W @ X + b  for q,k,v  (blockIdx.z selects which)
//   Q,K: [h][n][64]   V: [h][n][80] (dims 0..63).  c = d*4 + h.
// ---------------------------------------------------------------------------
__global__ void proj_kernel(const float* __restrict__ Xq, const float* __restrict__ Xk,
                            const float* __restrict__ Xv,
                            const float* __restrict__ Wq, const float* __restrict__ Wk,
                            const float* __restrict__ Wv,
                            const float* __restrict__ bq, const float* __restrict__ bk,
                            const float* __restrict__ bv,
                            float* __restrict__ Qo, float* __restrict__ Ko,
                            float* __restrict__ Vo) {
  const int lane = threadIdx.x & 31;
  const int col  = lane & 15;
  const int half = lane >> 4;
  const int nbase = blockIdx.x * 16;
  const int cbase = blockIdx.y * 16;
  const int which = blockIdx.z;

  const float* X = (which == 0) ? Xq : (which == 1) ? Xk : Xv;
  const float* W = (which == 0) ? Wq : (which == 1) ? Wk : Wv;
  const float* b = (which == 0) ? bq : (which == 1) ? bk : bv;
  float*       O = (which == 0) ? Qo : (which == 1) ? Ko : Vo;
  const int ostride = (which == 2) ? VDIM : DHEAD;

  const int m = cbase + col;  // A-matrix row (output channel)
  v8f acc = {0.f, 0.f, 0.f, 0.f, 0.f, 0.f, 0.f, 0.f};
  for (int k0 = 0; k0 < DM; k0 += 4) {
    v2f a = *(const v2f*)&W[(size_t)m * DM + k0 + 2 * half];
    v2f bb;
    bb.x = X[(size_t)(k0 + 2 * half + 0) * N_TOK + nbase + col];
    bb.y = X[(size_t)(k0 + 2 * half + 1) * N_TOK + nbase + col];
    acc = wmma4(a, bb, acc);
  }
#pragma unroll
  for (int j = 0; j < 8; ++j) {
    const int c = cbase + j + 8 * half;
    const int n = nbase + col;
    const int h = c & 3;
    const int d = c >> 2;
    O[((size_t)h * N_TOK + n) * ostride + d] = acc[j] + b[c];
  }
}

// ---------------------------------------------------------------------------
// Kernel 2: fill V augmentation: dims 64..66 = coords1, 67..79 = 0
// ---------------------------------------------------------------------------
__global__ void vaug_kernel(const float* __restrict__ coords1, float* __restrict__ Vo) {
  const int idx = blockIdx.x * blockDim.x + threadIdx.x;
  if (idx >= NH * N_TOK) return;
  const int h = idx >> 12;
  const int n = idx & (N_TOK - 1);
  float* row = Vo + ((size_t)h * N_TOK + n) * VDIM;
  row[64] = coords1[0 * N_TOK + n];
  row[65] = coords1[1 * N_TOK + n];
  row[66] = coords1[2 * N_TOK + n];
#pragma unroll
  for (int d = 67; d < 80; ++d) row[d] = 0.f;
}

// ---------------------------------------------------------------------------
// Kernel 3: flash attention, 4 waves/block sharing async-staged K/V tiles.
// Each wave owns a 16-query tile; per iteration stage 32 keys (K: 8KB, V: 10KB)
// into double-buffered LDS with global_load_async_to_lds_b128, overlap the
// next block's DMA with this block's 72 WMMAs.
// ---------------------------------------------------------------------------
__global__ void __launch_bounds__(128)
flash_kernel(const float* __restrict__ Qws, const float* __restrict__ Kws,
             const float* __restrict__ Vws, float* __restrict__ Y) {
  __shared__ float Ks[2][KB * DHEAD];  // 2 x 8 KB
  __shared__ float Vs[2][KB * VDIM];   // 2 x 10 KB
  __shared__ float Pt[4][16 * KB];     // per-wave P tiles (C->A layout hop)

  const int tid  = threadIdx.x;
  const int wave = tid >> 5;
  const int lane = tid & 31;
  const int col  = lane & 15;
  const int half = lane >> 4;
  const int h     = blockIdx.y;
  const int qbase = (blockIdx.x * 4 + wave) * 16;

  const float* Qh = Qws + (size_t)h * N_TOK * DHEAD;
  const float* Kh = Kws + (size_t)h * N_TOK * DHEAD;
  const float* Vh = Vws + (size_t)h * N_TOK * VDIM;
  float* Pw = &Pt[wave][0];

  // stage one 32-key block of K and V into LDS buffer `buf` (async, 16B/lane)
  auto stage = [&](int buf, int kb) {
    const char* kg = (const char*)(Kh + (size_t)kb * DHEAD);   // contiguous 8KB
    const unsigned ko = lds_off(&Ks[buf][0]);
#pragma unroll
    for (int i = 0; i < 4; ++i) {
      const int c = i * 128 + tid;
      async_ld_b128(ko + c * 16, kg + c * 16);
    }
    const char* vg = (const char*)(Vh + (size_t)kb * VDIM);    // contiguous 10KB
    const unsigned vo = lds_off(&Vs[buf][0]);
#pragma unroll
    for (int i = 0; i < 5; ++i) {
      const int c = i * 128 + tid;
      async_ld_b128(vo + c * 16, vg + c * 16);
    }
  };

  stage(0, 0);  // prologue DMA overlaps Q-fragment loads below

  // Q fragments for all 16 K-steps of d, pre-scaled by 1/sqrt(64) = 0.125
  v2f qf[16];
#pragma unroll
  for (int s = 0; s < 16; ++s) {
    v2f t = *(const v2f*)&Qh[(size_t)(qbase + col) * DHEAD + 4 * s + 2 * half];
    qf[s].x = t.x * 0.125f;
    qf[s].y = t.y * 0.125f;
  }

  v8f acc[5];
#pragma unroll
  for (int t = 0; t < 5; ++t) acc[t] = (v8f){0.f, 0.f, 0.f, 0.f, 0.f, 0.f, 0.f, 0.f};
  float mrow[8], lrow[8];
#pragma unroll
  for (int j = 0; j < 8; ++j) { mrow[j] = -INFINITY; lrow[j] = 0.f; }

  for (int kb = 0; kb < N_TOK; kb += KB) {
    const int buf = (kb / KB) & 1;
    wait_async0();     // this buffer's DMA done (issued last iteration)
    __syncthreads();   // visible to all 4 waves
    if (kb + KB < N_TOK) stage(buf ^ 1, kb + KB);  // overlap next DMA

    const float* Kb = &Ks[buf][0];
    const float* Vb = &Vs[buf][0];

    // ---- S tiles: 16 queries x (2 x 16 keys) ----------------------------
    v8f S0 = {0.f, 0.f, 0.f, 0.f, 0.f, 0.f, 0.f, 0.f};
    v8f S1 = {0.f, 0.f, 0.f, 0.f, 0.f, 0.f, 0.f, 0.f};
#pragma unroll
    for (int s = 0; s < 16; ++s) {
      v2f b0 = *(const v2f*)&Kb[(size_t)col * DHEAD + 4 * s + 2 * half];
      S0 = wmma4(qf[s], b0, S0);
    }
#pragma unroll
    for (int s = 0; s < 16; ++s) {
      v2f b1 = *(const v2f*)&Kb[(size_t)(16 + col) * DHEAD + 4 * s + 2 * half];
      S1 = wmma4(qf[s], b1, S1);
    }

    // ---- online softmax over 32 keys (rows live on one VGPR per half) ---
#pragma unroll
    for (int j = 0; j < 8; ++j) {
      float t = fmaxf(S0[j], S1[j]);
#pragma unroll
      for (int m = 1; m < 16; m <<= 1) t = fmaxf(t, __shfl_xor(t, m, 32));
      const float mnew  = fmaxf(mrow[j], t);
      const float alpha = __expf(mrow[j] - mnew);
      mrow[j] = mnew;
      const float p0 = __expf(S0[j] - mnew);
      const float p1 = __expf(S1[j] - mnew);
      float rs = p0 + p1;
#pragma unroll
      for (int m = 1; m < 16; m <<= 1) rs += __shfl_xor(rs, m, 32);
      lrow[j] = lrow[j] * alpha + rs;
#pragma unroll
      for (int t5 = 0; t5 < 5; ++t5) acc[t5][j] *= alpha;
      Pw[(j + 8 * half) * KB + col]      = p0;  // C layout -> LDS
      Pw[(j + 8 * half) * KB + 16 + col] = p1;
    }

    // ---- reload P in A layout (row = query, k = key) --------------------
    v2f af[8];
#pragma unroll
    for (int s = 0; s < 8; ++s)
      af[s] = *(const v2f*)&Pw[col * KB + 4 * s + 2 * half];

    // ---- acc += P * V  (5 dim-tiles of 16) ------------------------------
#pragma unroll
    for (int s = 0; s < 8; ++s) {
      const float* vp0 = &Vb[(size_t)(4 * s + 2 * half) * VDIM + col];
      const float* vp1 = vp0 + VDIM;
#pragma unroll
      for (int t5 = 0; t5 < 5; ++t5) {
        v2f bv;
        bv.x = vp0[t5 * 16];
        bv.y = vp1[t5 * 16];
        acc[t5] = wmma4(af[s], bv, acc[t5]);
      }
    }
    __syncthreads();  // all waves done with this buffer before it's re-DMA'd
  }

  // ---- epilogue: x = acc / l, scatter into y channels d*4+h --------------
#pragma unroll
  for (int j = 0; j < 8; ++j) {
    const float inv = 1.f / lrow[j];
    const int q = qbase + j + 8 * half;
#pragma unroll
    for (int t5 = 0; t5 < 5; ++t5) {
      const int d = t5 * 16 + col;
      if (d < 67) {
        Y[(size_t)(d * 4 + h) * N_TOK + q] = acc[t5][j] * inv;
      }
    }
  }
}

// ---------------------------------------------------------------------------
// Kernel 4: augmentation channels: aug1 = x[64..66]-coords0 -> ch (67+j)*4+h
//           aug2 = ||aug1||                                 -> ch 280+h
// ---------------------------------------------------------------------------
__global__ void aug_kernel(const float* __restrict__ coords0, float* __restrict__ Y) {
  const int idx = blockIdx.x * blockDim.x + threadIdx.x;
  if (idx >= NH * N_TOK) return;
  const int h = idx >> 12;
  const int n = idx & (N_TOK - 1);
  const float a0 = Y[(size_t)(64 * 4 + h) * N_TOK + n] - coords0[0 * N_TOK + n];
  const float a1 = Y[(size_t)(65 * 4 + h) * N_TOK + n] - coords0[1 * N_TOK + n];
  const float a2 = Y[(size_t)(66 * 4 + h) * N_TOK + n] - coords0[2 * N_TOK + n];
  Y[(size_t)(67 * 4 + h) * N_TOK + n] = a0;
  Y[(size_t)(68 * 4 + h) * N_TOK + n] = a1;
  Y[(size_t)(69 * 4 + h) * N_TOK + n] = a2;
  Y[(size_t)(70 * 4 + h) * N_TOK + n] = sqrtf(a0 * a0 + a1 * a1 + a2 * a2);
}

// ---------------------------------------------------------------------------
// Kernel 5: out = Wm[284x284] @ y[284x4096] + bm.  M padded to 288 (18 tiles),
// clamped A loads + guarded stores.  One wave per 16x16 tile, K = 284 = 71*4.
// ---------------------------------------------------------------------------
__global__ void out_gemm_kernel(const float* __restrict__ Wm, const float* __restrict__ bm,
                                const float* __restrict__ Y, float* __restrict__ Out) {
  const int lane = threadIdx.x & 31;
  const int col  = lane & 15;
  const int half = lane >> 4;
  const int nbase = blockIdx.x * 16;
  const int mbase = blockIdx.y * 16;

  int m = mbase + col;
  if (m > DOUT - 1) m = DOUT - 1;  // clamp pad rows (stores are guarded)

  v8f acc = {0.f, 0.f, 0.f, 0.f, 0.f, 0.f, 0.f, 0.f};
  for (int k0 = 0; k0 < DOUT; k0 += 4) {
    v2f a = *(const v2f*)&Wm[(size_t)m * DOUT + k0 + 2 * half];
    v2f b;
    b.x = Y[(size_t)(k0 + 2 * half + 0) * N_TOK + nbase + col];
    b.y = Y[(size_t)(k0 + 2 * half + 1) * N_TOK + nbase + col];
    acc = wmma4(a, b, acc);
  }
#pragma unroll
  for (int j = 0; j < 8; ++j) {
    const int mm = mbase + j + 8 * half;
    if (mm < DOUT) Out[(size_t)mm * N_TOK + nbase + col] = acc[j] + bm[mm];
  }
}

// ---------------------------------------------------------------------------
extern "C" void kernel_launch(void* const* d_in, const int* in_sizes, int n_in,
                              void* d_out, int out_size, void* d_ws, size_t ws_size,
                              hipStream_t stream) {
  const float* query   = (const float*)d_in[0];
  const float* key     = (const float*)d_in[1];
  const float* value   = (const float*)d_in[2];
  const float* coords0 = (const float*)d_in[3];
  const float* coords1 = (const float*)d_in[4];
  const float* Wq = (const float*)d_in[5];
  const float* bq = (const float*)d_in[6];
  const float* Wk = (const float*)d_in[7];
  const float* bk = (const float*)d_in[8];
  const float* Wv = (const float*)d_in[9];
  const float* bv = (const float*)d_in[10];
  const float* Wm = (const float*)d_in[11];
  const float* bm = (const float*)d_in[12];

  float* ws = (float*)d_ws;
  float* Q = ws;                                  // [4][4096][64]
  float* K = Q + (size_t)NH * N_TOK * DHEAD;      // [4][4096][64]
  float* V = K + (size_t)NH * N_TOK * DHEAD;      // [4][4096][80]
  float* Y = V + (size_t)NH * N_TOK * VDIM;       // [284][4096]

  proj_kernel<<<dim3(N_TOK / 16, DM / 16, 3), dim3(32), 0, stream>>>(
      query, key, value, Wq, Wk, Wv, bq, bk, bv, Q, K, V);
  vaug_kernel<<<dim3((NH * N_TOK) / 256), dim3(256), 0, stream>>>(coords1, V);
  flash_kernel<<<dim3(N_TOK / 64, NH), dim3(128), 0, stream>>>(Q, K, V, Y);
  aug_kernel<<<dim3((NH * N_TOK) / 256), dim3(256), 0, stream>>>(coords0, Y);
  out_gemm_kernel<<<dim3(N_TOK / 16, 18), dim3(32), 0, stream>>>(Wm, bm, Y, (float*)d_out);
}